// KGCN_75608604279516
// MI455X (gfx1250) — compile-verified
//
#include <hip/hip_runtime.h>
#include <math.h>

typedef __attribute__((ext_vector_type(2))) float v2f;
typedef __attribute__((ext_vector_type(8))) float v8f;
typedef __attribute__((ext_vector_type(4))) int   v4i;

#define DIM 32
#define NN  4
#define BT  16     // batch elements per block
#define NT  256    // threads per block (8 wave32)

#define XS  68     // padded row stride for [.|.] concat matrices (bank-conflict-free A loads)
#define WTS 68     // padded row stride for transposed W in LDS
#define H1S 33     // padded row stride for h1 / out tiles
#define EVS 36     // padded row stride for ev2 staging (conflict-free B128 fills)

#if defined(__gfx1250__) && __has_builtin(__builtin_amdgcn_global_load_async_to_lds_b128)
#define HAS_ASYNC_LDS 1
#else
#define HAS_ASYNC_LDS 0
#endif

typedef __attribute__((address_space(1))) v4i* gv4p;   // global 16B chunk
typedef __attribute__((address_space(3))) v4i* lv4p;   // LDS    16B chunk

// Copy 16 bytes global -> LDS. Async (ASYNCcnt-tracked) when the gfx1250
// builtin is available, synchronous fallback otherwise.
__device__ __forceinline__ void row_chunk16(const float* __restrict__ g, float* l) {
#if HAS_ASYNC_LDS
  __builtin_amdgcn_global_load_async_to_lds_b128((gv4p)g, (lv4p)l, 0, 0);
#else
  l[0] = g[0]; l[1] = g[1]; l[2] = g[2]; l[3] = g[3];
#endif
}

__device__ __forceinline__ void wait_async_copies() {
#if HAS_ASYNC_LDS
#if __has_builtin(__builtin_amdgcn_s_wait_asynccnt)
  __builtin_amdgcn_s_wait_asynccnt(0);
#else
  asm volatile("s_wait_asynccnt 0" ::: "memory");
#endif
#endif
}

// [16 x 64] (LDS, row stride XS) @ [64 x 32] (LDS, transposed Wt[n][k], stride WTS)
// via V_WMMA_F32_16X16X4_F32; two 16-col halves accumulated in acc0/acc1.
// A layout (32-bit 16x4): lane m<16 holds (M=m, K=k,k+1); lane 16+m holds (M=m, K=k+2,k+3).
// B layout (32-bit 4x16): lane n<16 holds (K=k,k+1, N=n); lane 16+n holds (K=k+2,k+3, N=n).
__device__ __forceinline__ void wmma_tile_16x64_64x32(
    const float* Arow, const float* Wt, v8f& acc0, v8f& acc1, int lane)
{
  const int m    = lane & 15;
  const int koff = (lane >> 4) << 1;   // 0 or 2
  #pragma unroll
  for (int k = 0; k < 64; k += 4) {
    v2f a, b0, b1v;
    a.x   = Arow[m * XS + k + koff];
    a.y   = Arow[m * XS + k + koff + 1];
    b0.x  = Wt[m * WTS + k + koff];
    b0.y  = Wt[m * WTS + k + koff + 1];
    b1v.x = Wt[(16 + m) * WTS + k + koff];
    b1v.y = Wt[(16 + m) * WTS + k + koff + 1];
    acc0 = __builtin_amdgcn_wmma_f32_16x16x4_f32(false, a, false, b0,  (short)0, acc0, false, false);
    acc1 = __builtin_amdgcn_wmma_f32_16x16x4_f32(false, a, false, b1v, (short)0, acc1, false, false);
  }
}

__global__ __launch_bounds__(NT)
void kgcn_kernel(const int*   __restrict__ user_indices,
                 const int*   __restrict__ item_indices,
                 const int*   __restrict__ adj_entity,
                 const int*   __restrict__ adj_relation,
                 const float* __restrict__ entity_emb,
                 const float* __restrict__ relation_emb,
                 const float* __restrict__ user_emb,
                 const float* __restrict__ W1, const float* __restrict__ b1,
                 const float* __restrict__ W2, const float* __restrict__ b2,
                 float* __restrict__ out, int B)
{
  __shared__ float rel_s[32 * DIM];         // relation table (4 KB)
  __shared__ float Wt1[32 * WTS];           // W1 transposed: Wt1[n][k]
  __shared__ float Wt2[32 * WTS];           // W2 transposed
  __shared__ float u_s  [BT * DIM];
  __shared__ float Xh0  [BT * XS];          // [ev0 | agg0]
  __shared__ float Xh1  [BT * NN * XS];     // [ev1 | agg1]
  __shared__ float Xout [BT * XS];          // [h0  | agg2]
  // time-sliced pool: phases 2-5 = ev2 staging (36 KB); phases 6+ = h1/out tiles
  __shared__ float pool [BT * NN * NN * EVS];
  float* ev2_s = pool;                      // [256][EVS]
  float* h1_s  = pool;                      // [64][H1S]
  float* out_s = pool + BT * NN * H1S;      // [16][H1S]
  __shared__ float attn0[BT * NN];
  __shared__ float attn1[BT * NN * NN];
  __shared__ int   e0_s[BT];
  __shared__ int   e1_s[BT * NN];
  __shared__ int   r0_s[BT * NN];
  __shared__ int   e2_s[BT * NN * NN];
  __shared__ int   r1_s[BT * NN * NN];

  const int t  = threadIdx.x;
  const int b0 = blockIdx.x * BT;

  // ---- phase 0: relation table, transposed weights, user rows, hop-0 indices
  for (int i = t; i < 32 * DIM; i += NT) rel_s[i] = relation_emb[i];
  for (int i = t; i < 64 * 32; i += NT) {   // W row-major [64][32] -> Wt[n][k]
    const int k = i >> 5, n = i & 31;
    Wt1[n * WTS + k] = W1[i];
    Wt2[n * WTS + k] = W2[i];
  }
  if (t < BT) e0_s[t] = item_indices[b0 + t];
  for (int i = t; i < BT * DIM; i += NT) {
    const int e = i >> 5, d = i & 31;
    u_s[e * DIM + d] = user_emb[(long)user_indices[b0 + e] * DIM + d];
  }
  if (t < BT * 8) {                         // ev0 rows -> Xh0[:,0:32] (async, needed phase 6)
    const int e = t >> 3, c = (t & 7) << 2;
    row_chunk16(&entity_emb[(long)item_indices[b0 + e] * DIM + c], &Xh0[e * XS + c]);
  }
  __syncthreads();

  // ---- phase 1: hop-1 adjacency
  if (t < BT * NN) {
    const long base = (long)e0_s[t >> 2] * NN + (t & 3);
    e1_s[t] = adj_entity  [base];
    r0_s[t] = adj_relation[base];
  }
  __syncthreads();

  // ---- phase 2: hop-2 adjacency; kick off all entity-row gathers async
  if (t < BT * 16) {
    const int  nk   = t & 15;
    const long base = (long)e1_s[(t >> 4) * NN + (nk >> 2)] * NN + (nk & 3);
    const int  e2i  = adj_entity[base];
    e2_s[t] = e2i;
    r1_s[t] = adj_relation[base];
    const float* grow = &entity_emb[(long)e2i * DIM];
    #pragma unroll
    for (int c = 0; c < DIM; c += 4)        // ev2 row -> staging (8 x B128 per row)
      row_chunk16(grow + c, &ev2_s[t * EVS + c]);
  }
  for (int i = t; i < BT * NN * 8; i += NT) { // ev1 rows -> Xh1[:,0:32]
    const int row = i >> 3, c = (i & 7) << 2;
    row_chunk16(&entity_emb[(long)e1_s[row] * DIM + c], &Xh1[row * XS + c]);
  }
  __syncthreads();   // publishes r1_s/e2_s; async row copies still in flight

  // ---- phase 3: attention scores = mean(u * rel) — overlaps the async gathers
  for (int i = t; i < BT * 16; i += NT) {
    const int    e  = i >> 4;
    const float* rv = &rel_s[r1_s[i] * DIM];
    float s = 0.f;
    #pragma unroll
    for (int d = 0; d < DIM; ++d) s += u_s[e * DIM + d] * rv[d];
    attn1[i] = s * (1.f / DIM);
  }
  if (t < BT * NN) {
    const int    e  = t >> 2;
    const float* rv = &rel_s[r0_s[t] * DIM];
    float s = 0.f;
    #pragma unroll
    for (int d = 0; d < DIM; ++d) s += u_s[e * DIM + d] * rv[d];
    attn0[t] = s * (1.f / DIM);
  }
  __syncthreads();

  // ---- phase 4: softmax over k = 4
  if (t < BT) {
    float* a = &attn0[t * 4];
    const float mx = fmaxf(fmaxf(a[0], a[1]), fmaxf(a[2], a[3]));
    const float x0 = __expf(a[0] - mx), x1 = __expf(a[1] - mx),
                x2 = __expf(a[2] - mx), x3 = __expf(a[3] - mx);
    const float inv = 1.f / (x0 + x1 + x2 + x3);
    a[0] = x0 * inv; a[1] = x1 * inv; a[2] = x2 * inv; a[3] = x3 * inv;
  } else if (t >= 64 && t < 64 + BT * NN) {
    float* a = &attn1[(t - 64) * 4];
    const float mx = fmaxf(fmaxf(a[0], a[1]), fmaxf(a[2], a[3]));
    const float x0 = __expf(a[0] - mx), x1 = __expf(a[1] - mx),
                x2 = __expf(a[2] - mx), x3 = __expf(a[3] - mx);
    const float inv = 1.f / (x0 + x1 + x2 + x3);
    a[0] = x0 * inv; a[1] = x1 * inv; a[2] = x2 * inv; a[3] = x3 * inv;
  }
  wait_async_copies();  // this wave's ev0/ev1/ev2 transfers landed in LDS
  __syncthreads();      // ... and every other wave passed its own wait

  // ---- phase 5: neighbor aggregation (all operands now in LDS)
  for (int i = t; i < BT * DIM; i += NT) {          // agg0 -> Xh0[:,32:]
    const int e = i >> 5, d = i & 31;
    float s = 0.f;
    #pragma unroll
    for (int j = 0; j < NN; ++j) s += attn0[e * 4 + j] * Xh1[(e * 4 + j) * XS + d];
    Xh0[e * XS + 32 + d] = s;
  }
  for (int i = t; i < BT * NN * DIM; i += NT) {     // agg1 -> Xh1[:,32:]
    const int row = i >> 5, d = i & 31;
    float s = 0.f;
    #pragma unroll
    for (int k = 0; k < NN; ++k)
      s += attn1[row * 4 + k] * ev2_s[(row * 4 + k) * EVS + d];
    Xh1[row * XS + 32 + d] = s;
  }
  __syncthreads();   // after this barrier the pool is reused as h1/out tiles

  // ---- phase 6: WMMA stage 1 — relu([X | agg] @ W1 + b1). 5 tiles of 16 rows.
  {
    const int wave = t >> 5, lane = t & 31;
    if (wave < 5) {
      const float* Arow = (wave < 4) ? &Xh1[wave * 16 * XS] : &Xh0[0];
      v8f acc0 = {}; v8f acc1 = {};
      wmma_tile_16x64_64x32(Arow, Wt1, acc0, acc1, lane);
      const int m = lane & 15;
      const int rbase = (lane >> 4) << 3;  // D layout: lanes>=16 hold rows 8..15
      #pragma unroll
      for (int v = 0; v < 8; ++v) {
        const int row = rbase + v;
        const float x0 = fmaxf(acc0[v] + b1[m],      0.f);
        const float x1 = fmaxf(acc1[v] + b1[16 + m], 0.f);
        if (wave < 4) {
          h1_s[(wave * 16 + row) * H1S + m]      = x0;
          h1_s[(wave * 16 + row) * H1S + 16 + m] = x1;
        } else {
          Xout[row * XS + m]      = x0;   // h0 is the self-half of Xout
          Xout[row * XS + 16 + m] = x1;
        }
      }
    }
  }
  __syncthreads();

  // ---- phase 7: agg2 over h1 with the hop-0 attention -> Xout[:,32:]
  for (int i = t; i < BT * DIM; i += NT) {
    const int e = i >> 5, d = i & 31;
    float s = 0.f;
    #pragma unroll
    for (int k = 0; k < NN; ++k) s += attn0[e * 4 + k] * h1_s[(e * 4 + k) * H1S + d];
    Xout[e * XS + 32 + d] = s;
  }
  __syncthreads();

  // ---- phase 8: WMMA stage 2 — tanh(Xout @ W2 + b2). 1 tile, wave 0.
  {
    const int wave = t >> 5, lane = t & 31;
    if (wave == 0) {
      v8f acc0 = {}; v8f acc1 = {};
      wmma_tile_16x64_64x32(&Xout[0], Wt2, acc0, acc1, lane);
      const int m = lane & 15;
      const int rbase = (lane >> 4) << 3;
      #pragma unroll
      for (int v = 0; v < 8; ++v) {
        const int row = rbase + v;
        out_s[row * H1S + m]      = tanhf(acc0[v] + b2[m]);
        out_s[row * H1S + 16 + m] = tanhf(acc1[v] + b2[16 + m]);
      }
    }
  }
  __syncthreads();

  // ---- phase 9: scores = sigmoid(u . out)
  if (t < BT && (b0 + t) < B) {
    float s = 0.f;
    #pragma unroll
    for (int d = 0; d < DIM; ++d) s += u_s[t * DIM + d] * out_s[t * H1S + d];
    out[b0 + t] = 1.f / (1.f + __expf(-s));
  }
}

extern "C" void kernel_launch(void* const* d_in, const int* in_sizes, int n_in,
                              void* d_out, int out_size, void* d_ws, size_t ws_size,
                              hipStream_t stream) {
  const int*   user_indices = (const int*)  d_in[0];
  const int*   item_indices = (const int*)  d_in[1];
  const int*   adj_entity   = (const int*)  d_in[2];
  const int*   adj_relation = (const int*)  d_in[3];
  const float* entity_emb   = (const float*)d_in[4];
  const float* relation_emb = (const float*)d_in[5];
  const float* user_emb     = (const float*)d_in[6];
  const float* W1           = (const float*)d_in[7];
  const float* b1           = (const float*)d_in[8];
  const float* W2           = (const float*)d_in[9];
  const float* b2           = (const float*)d_in[10];
  float* out = (float*)d_out;

  const int B    = in_sizes[0];
  const int grid = B / BT;   // B = 65536 -> 4096 blocks
  hipLaunchKernelGGL(kgcn_kernel, dim3(grid), dim3(NT), 0, stream,
                     user_indices, item_indices, adj_entity, adj_relation,
                     entity_emb, relation_emb, user_emb, W1, b1, W2, b2, out, B);
}